// DSGNNEncoder_7713761264089
// MI455X (gfx1250) — compile-verified
//
#include <hip/hip_runtime.h>
#include <math.h>

// ---------------------------------------------------------------------------
// GAT 2-layer forward for MI455X (gfx1250, wave32).
//   h = x @ W            -> V_WMMA_F32_16X16X4_F32, 16x128 tile per wave
//                           (A fragment loaded once, feeds 8 WMMAs)
//   a_src/a_dst dots     -> wave32 shuffle reductions
//   segment softmax      -> atomic f32 max (sign trick) + atomic exp-sum
//   aggregation          -> wave-per-edge coalesced global_atomic_add_f32
// Working set (~160 MB) is resident in the 192 MB L2; edge scatter dominates.
// ---------------------------------------------------------------------------

typedef __attribute__((ext_vector_type(2))) float v2f;
typedef __attribute__((ext_vector_type(8))) float v8f;

#define D_FEAT    128
#define HEADS     4
#define C_OUT     32
#define NEG_SLOPE 0.2f

// ---------------- GEMM: H[N,128] = X[N,128] @ W[128,128] --------------------
// One wave computes a 16x128 output slab: 8 f32 accumulators, A fragment
// reused across the 8 column tiles. Early exit is wave-uniform, so EXEC is
// all-ones for every executed WMMA (ISA requirement).
__global__ __launch_bounds__(256) void gat_gemm(const float* __restrict__ X,
                                                const float* __restrict__ W,
                                                float* __restrict__ H,
                                                int nRowTiles)
{
    const int wave = blockIdx.x * 8 + (threadIdx.x >> 5);
    if (wave >= nRowTiles) return;             // uniform per-wave exit
    const int lane = threadIdx.x & 31;
    const int half = lane >> 4;                // 0: lanes 0-15, 1: lanes 16-31
    const int lr   = lane & 15;
    const int rowBase = wave * 16;

    const float* __restrict__ arow = X + (size_t)(rowBase + lr) * D_FEAT;

    v8f c[8];
    const v8f zero = {};
#pragma unroll
    for (int t = 0; t < 8; ++t) c[t] = zero;

#pragma unroll 4
    for (int k0 = 0; k0 < D_FEAT; k0 += 4) {
        const int ka = k0 + 2 * half;          // A 16x4 layout: K = 2*half + v
        v2f a;
        a.x = arow[ka];
        a.y = arow[ka + 1];
        const float* __restrict__ wr0 = W + (size_t)ka * D_FEAT + lr;
        const float* __restrict__ wr1 = wr0 + D_FEAT;
#pragma unroll
        for (int t = 0; t < 8; ++t) {          // B 4x16 layout per column tile
            v2f b;
            b.x = wr0[t * 16];
            b.y = wr1[t * 16];
            c[t] = __builtin_amdgcn_wmma_f32_16x16x4_f32(
                       /*neg_a=*/false, a, /*neg_b=*/false, b,
                       /*c_mod=*/(short)0, c[t],
                       /*reuse_a=*/false, /*reuse_b=*/false);
        }
    }

#pragma unroll
    for (int t = 0; t < 8; ++t) {
        float* __restrict__ out = H + (size_t)rowBase * D_FEAT + t * 16 + lr;
#pragma unroll
        for (int r = 0; r < 8; ++r)            // C/D: VGPR r -> M = r + 8*half
            out[(size_t)(r + 8 * half) * D_FEAT] = c[t][r];
    }
}

// ------------- per-node attention dots: a[n,h] = <h[n,h,:], att[h,:]> -------
__global__ __launch_bounds__(128) void gat_att_dots(const float* __restrict__ H,
                                                    const float* __restrict__ attS,
                                                    const float* __restrict__ attD,
                                                    float* __restrict__ aS,
                                                    float* __restrict__ aD)
{
    const int node = blockIdx.x * 4 + (threadIdx.x >> 5);   // wave per node
    const int lane = threadIdx.x & 31;
    const float* __restrict__ hv = H + (size_t)node * D_FEAT;
#pragma unroll
    for (int hd = 0; hd < HEADS; ++hd) {
        const float x = hv[hd * C_OUT + lane];
        float s = x * attS[hd * C_OUT + lane];
        float d = x * attD[hd * C_OUT + lane];
#pragma unroll
        for (int off = 16; off > 0; off >>= 1) {
            s += __shfl_xor(s, off, 32);
            d += __shfl_xor(d, off, 32);
        }
        if (lane == 0) {
            aS[node * HEADS + hd] = s;
            aD[node * HEADS + hd] = d;
        }
    }
}

// ---------------------------- utility fill ---------------------------------
__global__ void fill_f32(float* __restrict__ p, float v, int n)
{
    const int t = blockIdx.x * blockDim.x + threadIdx.x;
    if (t < n) p[t] = v;
}

// -------- float atomic max via monotone int/uint bit trick ------------------
__device__ __forceinline__ void atomicMaxF(float* addr, float val)
{
    if (val >= 0.0f)
        atomicMax((int*)addr, __float_as_int(val));
    else
        atomicMin((unsigned int*)addr, __float_as_uint(val));
}

__device__ __forceinline__ float lrelu(float v)
{
    return v > 0.0f ? v : NEG_SLOPE * v;
}

// edge id >= nE encodes the self loop (src = dst = e - nE)
__device__ __forceinline__ void edge_endpoints(const long long* __restrict__ src,
                                               const long long* __restrict__ dst,
                                               int e, int nE, int& s, int& d)
{
    if (e < nE) { s = (int)src[e]; d = (int)dst[e]; }
    else        { s = d = e - nE; }
}

// pass 1: segment max over destinations
__global__ __launch_bounds__(256) void edge_max(const long long* __restrict__ src,
                                                const long long* __restrict__ dst,
                                                const float* __restrict__ aS,
                                                const float* __restrict__ aD,
                                                float* __restrict__ m,
                                                int nE, int nTot)
{
    const int t = blockIdx.x * blockDim.x + threadIdx.x;
    if (t >= nTot * HEADS) return;
    const int e = t >> 2, hd = t & 3;
    int s, d;
    edge_endpoints(src, dst, e, nE, s, d);
    const float ev = lrelu(aS[s * HEADS + hd] + aD[d * HEADS + hd]);
    atomicMaxF(&m[d * HEADS + hd], ev);
}

// pass 2: segment sum of exp(e - max)
__global__ __launch_bounds__(256) void edge_expsum(const long long* __restrict__ src,
                                                   const long long* __restrict__ dst,
                                                   const float* __restrict__ aS,
                                                   const float* __restrict__ aD,
                                                   const float* __restrict__ m,
                                                   float* __restrict__ denom,
                                                   int nE, int nTot)
{
    const int t = blockIdx.x * blockDim.x + threadIdx.x;
    if (t >= nTot * HEADS) return;
    const int e = t >> 2, hd = t & 3;
    int s, d;
    edge_endpoints(src, dst, e, nE, s, d);
    const float ev = lrelu(aS[s * HEADS + hd] + aD[d * HEADS + hd]);
    atomicAdd(&denom[d * HEADS + hd], expf(ev - m[d * HEADS + hd]));
}

// pass 3: acc[dst] += alpha * h[src] ; wave per edge, lane = channel in head
__global__ __launch_bounds__(256) void edge_scatter(const long long* __restrict__ src,
                                                    const long long* __restrict__ dst,
                                                    const float* __restrict__ aS,
                                                    const float* __restrict__ aD,
                                                    const float* __restrict__ m,
                                                    const float* __restrict__ denom,
                                                    const float* __restrict__ H,
                                                    float* __restrict__ acc,
                                                    int nE, int nTot)
{
    const int wave = blockIdx.x * 8 + (threadIdx.x >> 5);
    if (wave >= nTot) return;
    const int lane = threadIdx.x & 31;
    int s, d;
    edge_endpoints(src, dst, wave, nE, s, d);
    const float* __restrict__ hs = H + (size_t)s * D_FEAT;
    float* __restrict__ ad = acc + (size_t)d * D_FEAT;
#pragma unroll
    for (int hd = 0; hd < HEADS; ++hd) {
        const float ev    = lrelu(aS[s * HEADS + hd] + aD[d * HEADS + hd]);
        const float alpha = expf(ev - m[d * HEADS + hd]) / denom[d * HEADS + hd];
        atomicAdd(&ad[hd * C_OUT + lane], alpha * hs[hd * C_OUT + lane]);
    }
}

// out = elu(acc + bias)
__global__ void bias_elu(const float* __restrict__ in, const float* __restrict__ b,
                         float* __restrict__ out, int n)
{
    const int t = blockIdx.x * blockDim.x + threadIdx.x;
    if (t < n) {
        const float v = in[t] + b[t & (D_FEAT - 1)];
        out[t] = v > 0.0f ? v : expm1f(v);
    }
}

// ---------------------------------------------------------------------------
static void run_layer(const float* in, const float* W, const float* attS,
                      const float* attD, const float* bias,
                      const long long* srcArr, const long long* dstArr,
                      float* h, float* aS, float* aD, float* mBuf, float* dBuf,
                      float* acc, float* out,
                      int N, int E, hipStream_t stream)
{
    const int ETOT      = E + N;
    const int nRowTiles = N / 16;
    const int nFeat     = N * D_FEAT;
    const int nNH       = N * HEADS;

    gat_gemm<<<(nRowTiles + 7) / 8, 256, 0, stream>>>(in, W, h, nRowTiles);
    gat_att_dots<<<N / 4, 128, 0, stream>>>(h, attS, attD, aS, aD);

    fill_f32<<<(nNH + 255) / 256, 256, 0, stream>>>(mBuf, -INFINITY, nNH);
    fill_f32<<<(nNH + 255) / 256, 256, 0, stream>>>(dBuf, 0.0f, nNH);
    fill_f32<<<(nFeat + 255) / 256, 256, 0, stream>>>(acc, 0.0f, nFeat);

    const int eThreads = ETOT * HEADS;
    edge_max<<<(eThreads + 255) / 256, 256, 0, stream>>>(srcArr, dstArr, aS, aD,
                                                         mBuf, E, ETOT);
    edge_expsum<<<(eThreads + 255) / 256, 256, 0, stream>>>(srcArr, dstArr, aS, aD,
                                                            mBuf, dBuf, E, ETOT);
    edge_scatter<<<(ETOT + 7) / 8, 256, 0, stream>>>(srcArr, dstArr, aS, aD,
                                                     mBuf, dBuf, h, acc, E, ETOT);
    bias_elu<<<(nFeat + 255) / 256, 256, 0, stream>>>(acc, bias, out, nFeat);
}

extern "C" void kernel_launch(void* const* d_in, const int* in_sizes, int n_in,
                              void* d_out, int out_size, void* d_ws, size_t ws_size,
                              hipStream_t stream)
{
    const float*     x    = (const float*)d_in[0];
    const long long* ei   = (const long long*)d_in[1];   // int64 [2, E]
    const float*     W1   = (const float*)d_in[2];
    const float*     as1  = (const float*)d_in[3];
    const float*     ad1  = (const float*)d_in[4];
    const float*     b1   = (const float*)d_in[5];
    const float*     W2   = (const float*)d_in[6];
    const float*     as2  = (const float*)d_in[7];
    const float*     ad2  = (const float*)d_in[8];
    const float*     b2   = (const float*)d_in[9];

    const int N = in_sizes[0] / D_FEAT;   // 100000 (multiple of 16)
    const int E = in_sizes[1] / 2;        // 1600000
    const long long* srcArr = ei;
    const long long* dstArr = ei + E;

    float* ws   = (float*)d_ws;
    float* h    = ws;                              // [N,128]
    float* x2   = h  + (size_t)N * D_FEAT;         // [N,128] layer-1 acc / layer-2 input
    float* aS   = x2 + (size_t)N * D_FEAT;         // [N,4]
    float* aD   = aS + (size_t)N * HEADS;          // [N,4]
    float* mBuf = aD + (size_t)N * HEADS;          // [N,4]
    float* dBuf = mBuf + (size_t)N * HEADS;        // [N,4]
    float* outF = (float*)d_out;                   // [N,128]

    // layer 1: x -> x2
    run_layer(x, W1, as1, ad1, b1, srcArr, dstArr,
              h, aS, aD, mBuf, dBuf, x2, x2, N, E, stream);
    // layer 2: x2 -> d_out (aggregate directly into output buffer)
    run_layer(x2, W2, as2, ad2, b2, srcArr, dstArr,
              h, aS, aD, mBuf, dBuf, outF, outF, N, E, stream);
}